// Autoencoder_76776835383930
// MI455X (gfx1250) — compile-verified
//
#include <hip/hip_runtime.h>
#include <hip/hip_bf16.h>

typedef __attribute__((ext_vector_type(2))) float v2f;
typedef __attribute__((ext_vector_type(4))) float v4f;
typedef __attribute__((ext_vector_type(8))) float v8f;
typedef __attribute__((ext_vector_type(4))) int   v4i;

// Problem constants (match reference)
constexpr int BATCH = 8192;
constexpr int D_IN  = 1024;
constexpr int HID   = 16384;
constexpr int D_OUT = 1024;
constexpr int K_TOP = 32;

// GEMM tiling
constexpr int TM = 128;            // batch rows per block
constexpr int TN = 128;            // hidden cols per block
constexpr int TK = 32;             // K stage
constexpr int LDS_PITCH = TK + 4;  // 36 dwords: keeps 16B align, spreads banks

#define GLOBAL_AS __attribute__((address_space(1)))
#define LDS_AS    __attribute__((address_space(3)))

#if defined(__has_builtin)
#if __has_builtin(__builtin_amdgcn_global_load_async_to_lds_b128)
#define HAVE_ASYNC_LDS 1
#endif
#if __has_builtin(__builtin_amdgcn_s_wait_asynccnt)
#define HAVE_WAIT_ASYNC 1
#endif
#endif

__device__ __forceinline__ void wait_async0() {
#if defined(HAVE_WAIT_ASYNC)
    __builtin_amdgcn_s_wait_asynccnt(0);
#else
    asm volatile("s_wait_asynccnt 0" ::: "memory");
#endif
}

// ---------------------------------------------------------------------------
// Encoder: pre = x @ W_enc^T + b_enc, ReLU.  f32 WMMA 16x16x4.
// Block = 256 threads = 8 wave32, waves arranged 4(M) x 2(N),
// each wave computes a 32x64 sub-tile = 2x4 grid of 16x16 WMMA tiles.
// Double-buffered LDS filled by async global->LDS DMA (ASYNCcnt path).
// ---------------------------------------------------------------------------
__global__ __launch_bounds__(256)
void enc_gemm_relu(const float* __restrict__ x,
                   const float* __restrict__ W,      // [HID, D_IN] row-major
                   const float* __restrict__ bias,   // [HID]
                   float* __restrict__ out)          // [BATCH, HID] dense
{
    __shared__ float sX[2][TM * LDS_PITCH];
    __shared__ float sW[2][TN * LDS_PITCH];

    const int tid   = threadIdx.x;
    const int lane  = tid & 31;
    const int wave  = tid >> 5;     // 0..7
    const int wm    = wave >> 1;    // 0..3  (M direction)
    const int wn    = wave & 1;     // 0..1  (N direction)
    const int l15   = lane & 15;
    const int lhalf = lane >> 4;    // 0/1

    const int row0 = blockIdx.y * TM;   // batch base
    const int col0 = blockIdx.x * TN;   // hidden base

    v8f acc[2][4];
#pragma unroll
    for (int a = 0; a < 2; ++a)
#pragma unroll
        for (int b = 0; b < 4; ++b)
#pragma unroll
            for (int r = 0; r < 8; ++r)
                acc[a][b][r] = 0.0f;

    // staging assignment: thread -> (row = tid/8 + i*32, quad-col = (tid%8)*4)
    const int lrow = tid >> 3;        // 0..31
    const int lq   = (tid & 7) * 4;   // 0,4,...,28

    // Issue one K-stage of async global->LDS b128 copies (16 KB per matrix).
    auto issue_stage = [&](int k0, int buf) {
#if defined(HAVE_ASYNC_LDS)
#pragma unroll
        for (int i = 0; i < 4; ++i) {
            const int r = lrow + i * 32;      // 0..127
            __builtin_amdgcn_global_load_async_to_lds_b128(
                (GLOBAL_AS v4i*)(x + (size_t)(row0 + r) * D_IN + k0 + lq),
                (LDS_AS v4i*)(&sX[buf][r * LDS_PITCH + lq]), 0, 0);
            __builtin_amdgcn_global_load_async_to_lds_b128(
                (GLOBAL_AS v4i*)(W + (size_t)(col0 + r) * D_IN + k0 + lq),
                (LDS_AS v4i*)(&sW[buf][r * LDS_PITCH + lq]), 0, 0);
        }
#else
        // Fallback: synchronous staging through VGPRs.
#pragma unroll
        for (int i = 0; i < 4; ++i) {
            const int r = lrow + i * 32;
            v4f vx = *(const v4f*)(x + (size_t)(row0 + r) * D_IN + k0 + lq);
            *(v4f*)(&sX[buf][r * LDS_PITCH + lq]) = vx;
            v4f vw = *(const v4f*)(W + (size_t)(col0 + r) * D_IN + k0 + lq);
            *(v4f*)(&sW[buf][r * LDS_PITCH + lq]) = vw;
        }
#endif
    };

    auto wait_stage = [&]() {
#if defined(HAVE_ASYNC_LDS)
        wait_async0();
#endif
        __syncthreads();
    };

    // Prologue: fill buffer 0.
    issue_stage(0, 0);
    wait_stage();

    int buf = 0;
    for (int k0 = 0; k0 < D_IN; k0 += TK, buf ^= 1) {
        const bool more = (k0 + TK) < D_IN;
        if (more)
            issue_stage(k0 + TK, buf ^ 1);   // overlap fill of next stage

        const float* __restrict__ cX = &sX[buf][0];
        const float* __restrict__ cW = &sW[buf][0];
#pragma unroll
        for (int ks = 0; ks < TK; ks += 4) {
            const int kk = ks + 2 * lhalf;    // even -> 8B aligned in LDS
            v2f afrag[2];
            v2f bfrag[4];
#pragma unroll
            for (int a = 0; a < 2; ++a)
                afrag[a] = *(const v2f*)(&cX[(wm * 32 + a * 16 + l15) * LDS_PITCH + kk]);
#pragma unroll
            for (int b = 0; b < 4; ++b)
                bfrag[b] = *(const v2f*)(&cW[(wn * 64 + b * 16 + l15) * LDS_PITCH + kk]);
#pragma unroll
            for (int a = 0; a < 2; ++a)
#pragma unroll
                for (int b = 0; b < 4; ++b)
                    acc[a][b] = __builtin_amdgcn_wmma_f32_16x16x4_f32(
                        false, afrag[a], false, bfrag[b],
                        (short)0, acc[a][b], false, false);
        }
        __syncthreads();          // all waves done reading `buf`
        if (more)
            wait_stage();         // all async fills of buf^1 landed, all waves past
    }

    // Epilogue: bias + ReLU, f32 store. C/D layout: vgpr r, lanes 0-15 -> M=r,
    // lanes 16-31 -> M=r+8; N = lane&15.
#pragma unroll
    for (int a = 0; a < 2; ++a) {
#pragma unroll
        for (int b = 0; b < 4; ++b) {
            const int col = col0 + wn * 64 + b * 16 + l15;
            const float bv = bias[col];
#pragma unroll
            for (int r = 0; r < 8; ++r) {
                const int row = row0 + wm * 32 + a * 16 + r + 8 * lhalf;
                float v = acc[a][b][r] + bv;
                v = v > 0.0f ? v : 0.0f;
                out[(size_t)row * HID + col] = v;
            }
        }
    }
}

// ---------------------------------------------------------------------------
// Per-row top-K via binary search on IEEE bit patterns (values are >= 0 after
// ReLU, so float order == uint order). Exactly K kept; ties broken by (index % 256, then index / 256).
// Rewrites the dense encoded row in place (non-top-K -> 0) and emits compact
// (val, idx) pairs for the sparse decoder.
// ---------------------------------------------------------------------------
__global__ __launch_bounds__(256)
void topk_select(float* __restrict__ enc,          // [BATCH, HID] in/out
                 float* __restrict__ vals,         // [BATCH, K_TOP]
                 int*   __restrict__ idxs)         // [BATCH, K_TOP]
{
    constexpr int TPB = 256;
    constexpr int EPT = HID / TPB;   // 64 elements per thread

    const int row = blockIdx.x;
    const int tid = threadIdx.x;
    float* rowp = enc + (size_t)row * HID;

    float    v[EPT];
    unsigned u[EPT];
#pragma unroll
    for (int j = 0; j < EPT; ++j) {
        v[j] = rowp[tid + j * TPB];
        u[j] = __float_as_uint(v[j]);
    }

    __shared__ int s_cnt;
    __shared__ int s_scan[TPB];

    // Binary search: smallest b with count(u > b) < K  ==> b == bits of K-th largest.
    unsigned lo = 0u, hi = 0x7F800000u;
    while (lo < hi) {
        const unsigned mid = lo + ((hi - lo) >> 1);
        int c = 0;
#pragma unroll
        for (int j = 0; j < EPT; ++j) c += (u[j] > mid) ? 1 : 0;
        for (int off = 16; off > 0; off >>= 1) c += __shfl_down(c, off, 32);
        if (tid == 0) s_cnt = 0;
        __syncthreads();
        if ((tid & 31) == 0) atomicAdd(&s_cnt, c);
        __syncthreads();
        const int total = s_cnt;
        __syncthreads();
        if (total >= K_TOP) lo = mid + 1; else hi = mid;
    }
    const unsigned thr = lo;

    // Per-thread counts of strictly-greater and equal elements.
    int gtL = 0, eqL = 0;
#pragma unroll
    for (int j = 0; j < EPT; ++j) {
        gtL += (u[j] > thr) ? 1 : 0;
        eqL += (u[j] == thr) ? 1 : 0;
    }

    // G = total strictly greater (< K by construction).
    {
        int c = gtL;
        for (int off = 16; off > 0; off >>= 1) c += __shfl_down(c, off, 32);
        if (tid == 0) s_cnt = 0;
        __syncthreads();
        if ((tid & 31) == 0) atomicAdd(&s_cnt, c);
        __syncthreads();
    }
    const int G = s_cnt;
    __syncthreads();

    // Exclusive block scans (Hillis-Steele) for output positions.
    auto scan_excl = [&](int val) -> int {
        s_scan[tid] = val;
        __syncthreads();
        for (int off = 1; off < TPB; off <<= 1) {
            const int t = (tid >= off) ? s_scan[tid - off] : 0;
            __syncthreads();
            s_scan[tid] += t;
            __syncthreads();
        }
        const int incl = s_scan[tid];
        __syncthreads();
        return incl - val;
    };
    const int gtPre = scan_excl(gtL);
    const int eqPre = scan_excl(eqL);

    const int need = K_TOP - G;   // equals to admit (order: tid, then j)
    int gp = gtPre;
    int ep = eqPre;

#pragma unroll
    for (int j = 0; j < EPT; ++j) {
        const int h = tid + j * TPB;
        const float val = v[j];
        bool keep = false;
        if (u[j] > thr) {
            const int p = gp++;
            vals[(size_t)row * K_TOP + p] = val;
            idxs[(size_t)row * K_TOP + p] = h;
            keep = true;
        } else if (u[j] == thr) {
            const int r = ep++;
            if (r < need) {
                const int p = G + r;
                vals[(size_t)row * K_TOP + p] = val;
                idxs[(size_t)row * K_TOP + p] = h;
                keep = true;
            }
        }
        rowp[h] = keep ? val : 0.0f;
    }
}

// ---------------------------------------------------------------------------
// Sparse decoder: decoded[b, o] = b_dec[o] + sum_k val[b,k] * W_dec[o, idx[b,k]]
// W_dec (67 MB) is L2-resident (192 MB L2), so the strided gathers are cheap.
// ---------------------------------------------------------------------------
__global__ __launch_bounds__(256)
void sparse_decode(const float* __restrict__ vals,
                   const int*   __restrict__ idxs,
                   const float* __restrict__ Wd,    // [D_OUT, HID] row-major
                   const float* __restrict__ bd,    // [D_OUT]
                   float* __restrict__ out)         // [BATCH, D_OUT]
{
    const int row = blockIdx.x;
    const int tid = threadIdx.x;

    __shared__ float sv[K_TOP];
    __shared__ int   si[K_TOP];
    if (tid < K_TOP) {
        sv[tid] = vals[(size_t)row * K_TOP + tid];
        si[tid] = idxs[(size_t)row * K_TOP + tid];
    }
    __syncthreads();

    for (int o = tid; o < D_OUT; o += 256) {
        float acc = bd[o];
#pragma unroll
        for (int k = 0; k < K_TOP; ++k)
            acc += sv[k] * Wd[(size_t)o * HID + si[k]];
        out[(size_t)row * D_OUT + o] = acc;
    }
}

// ---------------------------------------------------------------------------
extern "C" void kernel_launch(void* const* d_in, const int* in_sizes, int n_in,
                              void* d_out, int out_size, void* d_ws, size_t ws_size,
                              hipStream_t stream) {
    const float* x     = (const float*)d_in[0];   // [B, D_IN]
    const float* W_enc = (const float*)d_in[1];   // [H, D_IN]
    const float* b_enc = (const float*)d_in[2];   // [H]
    const float* W_dec = (const float*)d_in[3];   // [D_OUT, H]
    const float* b_dec = (const float*)d_in[4];   // [D_OUT]
    // d_in[5] = topk scalar (compile-time K_TOP here)

    float* out_enc = (float*)d_out;                       // [B, H]
    float* out_dec = out_enc + (size_t)BATCH * HID;       // [B, D_OUT]

    float* ws_vals = (float*)d_ws;                        // [B, K]
    int*   ws_idx  = (int*)(ws_vals + (size_t)BATCH * K_TOP);

    dim3 gGemm(HID / TN, BATCH / TM);   // (128, 64)
    enc_gemm_relu<<<gGemm, 256, 0, stream>>>(x, W_enc, b_enc, out_enc);
    topk_select<<<BATCH, 256, 0, stream>>>(out_enc, ws_vals, ws_idx);
    sparse_decode<<<BATCH, 256, 0, stream>>>(ws_vals, ws_idx, W_dec, b_dec, out_dec);
}